// cross_block_14542759264790
// MI455X (gfx1250) — compile-verified
//
#include <hip/hip_runtime.h>

// MI455X / gfx1250, wave32. All matrix math through V_WMMA_F32_16X16X4_F32
// (fp32 in, fp32 accumulate). After re-association the op is ~8.6 GFLOP over a
// ~36 MB L2-resident working set -> bandwidth/latency bound, so full-precision
// WMMA is free and matches the fp32 reference.

typedef float v2f __attribute__((ext_vector_type(2)));
typedef float v8f __attribute__((ext_vector_type(8)));

__device__ __forceinline__ v8f wmma4(v2f a, v2f b, v8f c) {
    // 8 args: (neg_a, A, neg_b, B, c_mod, C, reuse_a, reuse_b)
    return __builtin_amdgcn_wmma_f32_16x16x4_f32(
        false, a, false, b, (short)0, c, false, false);
}

// ---------------------------------------------------------------------------
// NT GEMM, 32x32 output tile per wave (2x2 blocks of 16x16 WMMA tiles):
//   C[m,n] = alpha * sum_k A[m*lda+k] * B[n*ldb+k]  (+bias[n]) (+res[m,n])
// Optional split-K: blockIdx.z = batch*kSplit + chunk, C indexed per-chunk.
// K/kSplit must be a multiple of 64; M,N multiples of 32.
//
// f32 WMMA fragment layout (ISA 7.12.2):
//   A (16x4): lane&15 = M;  a.x -> K=2*(lane>>4), a.y -> K=2*(lane>>4)+1
//   B (4x16): lane&15 = N;  b.x -> K=2*(lane>>4), b.y -> K=2*(lane>>4)+1
//   C (16x16): lane&15 = N; acc[v] -> M = v + 8*(lane>>4)
// Both fragments are K-contiguous pairs -> single global_load_b64 each, so a
// k-step is 4 b64 loads feeding 4 WMMAs (2x reuse of each fragment).
// ---------------------------------------------------------------------------
__global__ __launch_bounds__(32)
void gemm_f32_nt32(const float* __restrict__ A, long strideAb, int lda,
                   const float* __restrict__ Bm, long strideBb, int ldb,
                   float* __restrict__ C, long strideCb, int ldc,
                   const float* __restrict__ bias,
                   const float* __restrict__ res, long strideRb,
                   int K, int kSplit, float alpha)
{
    const int lane = threadIdx.x & 31;
    const int half = lane >> 4;
    const int row  = lane & 15;
    const int j0 = blockIdx.x << 5;   // N tile base
    const int i0 = blockIdx.y << 5;   // M tile base
    const int bz = blockIdx.z;
    const int b     = bz / kSplit;
    const int chunk = bz - b * kSplit;
    const int kc    = K / kSplit;

    A  += (long)b * strideAb;
    Bm += (long)b * strideBb;
    C  += (long)bz * strideCb;        // per-(batch,chunk) slot when split
    if (res) res += (long)b * strideRb;

    const float* ap0 = A  + (long)(i0 + row) * lda + 2 * half + (long)chunk * kc;
    const float* ap1 = ap0 + (long)16 * lda;
    const float* bp0 = Bm + (long)(j0 + row) * ldb + 2 * half + (long)chunk * kc;
    const float* bp1 = bp0 + (long)16 * ldb;

    v8f acc[2][2];
    const v8f zero = {};
    acc[0][0] = zero; acc[0][1] = zero; acc[1][0] = zero; acc[1][1] = zero;

    for (int kk = 0; kk < kc; kk += 64) {
        // keep the four k-streams one 256B block ahead (global_prefetch_b8)
        __builtin_prefetch(ap0 + kk + 64, 0, 3);
        __builtin_prefetch(ap1 + kk + 64, 0, 3);
        __builtin_prefetch(bp0 + kk + 64, 0, 3);
        __builtin_prefetch(bp1 + kk + 64, 0, 3);
#pragma unroll 8
        for (int k0 = kk; k0 < kk + 64; k0 += 4) {
            v2f a0 = *(const v2f*)(ap0 + k0);
            v2f a1 = *(const v2f*)(ap1 + k0);
            v2f b0 = *(const v2f*)(bp0 + k0);
            v2f b1 = *(const v2f*)(bp1 + k0);
            acc[0][0] = wmma4(a0, b0, acc[0][0]);
            acc[0][1] = wmma4(a0, b1, acc[0][1]);
            acc[1][0] = wmma4(a1, b0, acc[1][0]);
            acc[1][1] = wmma4(a1, b1, acc[1][1]);
        }
    }

#pragma unroll
    for (int bi = 0; bi < 2; ++bi) {
#pragma unroll
        for (int bj = 0; bj < 2; ++bj) {
            const int n = j0 + 16 * bj + row;
            const float bn = bias ? bias[n] : 0.0f;
#pragma unroll
            for (int v = 0; v < 8; ++v) {
                const int m = i0 + 16 * bi + v + 8 * half;
                float out = acc[bi][bj][v] * alpha + bn;
                if (res) out += res[(long)m * ldc + n];
                C[(long)m * ldc + n] = out;
            }
        }
    }
}

// ---------------------------------------------------------------------------
// TN GEMM, 32x32 tile per wave: C[i,j] = alpha * sum_k A[k*lda+i] * B[k*ldb+j]
// (computes S^T @ X with S,X row-major). K-strided scalar loads, coalesced
// across lanes (16 contiguous floats per half-wave).
// ---------------------------------------------------------------------------
__global__ __launch_bounds__(32)
void gemm_f32_tn32(const float* __restrict__ A, long strideAb, int lda,
                   const float* __restrict__ Bm, long strideBb, int ldb,
                   float* __restrict__ C, long strideCb, int ldc,
                   int K, float alpha)
{
    const int lane = threadIdx.x & 31;
    const int half = lane >> 4;
    const int row  = lane & 15;
    const int j0 = blockIdx.x << 5;
    const int i0 = blockIdx.y << 5;
    const int b  = blockIdx.z;

    const float* ap = A  + (long)b * strideAb + (long)(2 * half) * lda + i0 + row;
    const float* bp = Bm + (long)b * strideBb + (long)(2 * half) * ldb + j0 + row;
    C += (long)b * strideCb;

    v8f acc[2][2];
    const v8f zero = {};
    acc[0][0] = zero; acc[0][1] = zero; acc[1][0] = zero; acc[1][1] = zero;

#pragma unroll 4
    for (int k0 = 0; k0 < K; k0 += 4) {
        const float* a_lo = ap + (long)k0 * lda;
        const float* b_lo = bp + (long)k0 * ldb;
        v2f a0, a1, b0, b1;
        a0.x = a_lo[0];        a0.y = a_lo[lda];
        a1.x = a_lo[16];       a1.y = a_lo[lda + 16];
        b0.x = b_lo[0];        b0.y = b_lo[ldb];
        b1.x = b_lo[16];       b1.y = b_lo[ldb + 16];
        acc[0][0] = wmma4(a0, b0, acc[0][0]);
        acc[0][1] = wmma4(a0, b1, acc[0][1]);
        acc[1][0] = wmma4(a1, b0, acc[1][0]);
        acc[1][1] = wmma4(a1, b1, acc[1][1]);
    }

#pragma unroll
    for (int bi = 0; bi < 2; ++bi) {
#pragma unroll
        for (int bj = 0; bj < 2; ++bj) {
            const int n = j0 + 16 * bj + row;
#pragma unroll
            for (int v = 0; v < 8; ++v) {
                const int m = i0 + 16 * bi + v + 8 * half;
                C[(long)m * ldc + n] = acc[bi][bj][v] * alpha;
            }
        }
    }
}

// ---------------------------------------------------------------------------
// Deterministic split-K combine: S[b,t] = alpha * sum_c P[(b*nChunk+c)*per + t]
// ---------------------------------------------------------------------------
__global__ __launch_bounds__(256)
void reduce_ksplit(const float* __restrict__ P, float* __restrict__ S,
                   int per, int nChunk, float alpha)
{
    const int idx = blockIdx.x * 256 + threadIdx.x;   // over Bn*per
    const int b = idx / per;
    const int t = idx - b * per;
    const float* p = P + (long)b * nChunk * per + t;
    float s = 0.0f;
    for (int c = 0; c < nChunk; ++c) s += p[(long)c * per];
    S[idx] = s * alpha;
}

// ---------------------------------------------------------------------------
// Host-side orchestration. All launches ordered on `stream`; deterministic.
// ---------------------------------------------------------------------------
extern "C" void kernel_launch(void* const* d_in, const int* in_sizes, int n_in,
                              void* d_out, int out_size, void* d_ws, size_t ws_size,
                              hipStream_t stream)
{
    (void)in_sizes; (void)n_in; (void)out_size; (void)ws_size;

    const float* detect = (const float*)d_in[0];
    const float* aim    = (const float*)d_in[1];
    const float* g_w  = (const float*)d_in[2];  const float* g_b  = (const float*)d_in[3];
    const float* g2_w = (const float*)d_in[4];  const float* g2_b = (const float*)d_in[5];
    const float* th_w = (const float*)d_in[6];  const float* th_b = (const float*)d_in[7];
    const float* ph_w = (const float*)d_in[8];  const float* ph_b = (const float*)d_in[9];
    const float* W_w  = (const float*)d_in[10]; const float* W_b  = (const float*)d_in[11];
    const float* Q_w  = (const float*)d_in[12]; const float* Q_b  = (const float*)d_in[13];

    const int  Bn = 4, N = 2048, C = 256, IC = 128, M2 = 4096;
    const int  KSPLIT = 8;                   // S-stage split-K factor
    const long actS = (long)N * C;           // per-batch activation stride (524288)
    const long PB   = (long)Bn * actS;       // whole-tensor float count (2097152)
    const long sS   = (long)IC * IC;         // 128x128 = 16384

    float* ws = (float*)d_ws;                // requires ~36.2 MB of workspace
    float* G    = ws + 0 * PB;               // detect @ g   ; later reused as NA
    float* G2   = ws + 1 * PB;               // aim    @ g2  ; later reused as ND
    float* TH   = ws + 2 * PB;               // aim    @ theta
    float* PH   = ws + 3 * PB;               // detect @ phi
    float* S1   = ws + 4 * PB;               // Bn x 128 x 128
    float* S2   = S1 + (long)Bn * sS;
    float* Part = S2 + (long)Bn * sS;        // Bn x KSPLIT x 128 x 128 partials
    float* NA = G;
    float* ND = G2;

    float* outDet = (float*)d_out;           // reference returns (non_det, non_aim)
    float* outAim = outDet + PB;

    const dim3 blk(32, 1, 1);
    const dim3 rblk(256, 1, 1);
    const dim3 gLin(C / 32, N / 32, Bn);          // 8 x 64 tiles
    const dim3 gS(IC / 32, IC / 32, Bn * KSPLIT); // 4 x 4 x 32 (split-K)
    const dim3 gBack(M2 / 32, IC / 32, Bn);       // 128 x 4 tiles
    const dim3 gRed((unsigned)(Bn * sS / 256), 1, 1);
    const float inv = 1.0f / 4096.0f;

    // Stage 1: four input projections, [2048,256] = X @ Wt^T + b   (NT, K=256)
    gemm_f32_nt32<<<gLin, blk, 0, stream>>>(detect, actS, C, g_w,  0, C, G,  actS, C, g_b,  nullptr, 0, C, 1, 1.0f);
    gemm_f32_nt32<<<gLin, blk, 0, stream>>>(aim,    actS, C, g2_w, 0, C, G2, actS, C, g2_b, nullptr, 0, C, 1, 1.0f);
    gemm_f32_nt32<<<gLin, blk, 0, stream>>>(aim,    actS, C, th_w, 0, C, TH, actS, C, th_b, nullptr, 0, C, 1, 1.0f);
    gemm_f32_nt32<<<gLin, blk, 0, stream>>>(detect, actS, C, ph_w, 0, C, PH, actS, C, ph_b, nullptr, 0, C, 1, 1.0f);

    // Stage 2: inner 128x128 correlations via split-K (associativity removes
    // the 4096x4096 f matrix entirely).
    // S1 = PH_r @ G_r^T / 4096 ; S2 = TH_r @ G2_r^T / 4096  (views as [128,4096])
    gemm_f32_nt32<<<gS, blk, 0, stream>>>(PH, actS, M2, G,  actS, M2, Part, sS, IC, nullptr, nullptr, 0, M2, KSPLIT, 1.0f);
    reduce_ksplit<<<gRed, rblk, 0, stream>>>(Part, S1, (int)sS, KSPLIT, inv);
    gemm_f32_nt32<<<gS, blk, 0, stream>>>(TH, actS, M2, G2, actS, M2, Part, sS, IC, nullptr, nullptr, 0, M2, KSPLIT, 1.0f);
    reduce_ksplit<<<gRed, rblk, 0, stream>>>(Part, S2, (int)sS, KSPLIT, inv);

    // Stage 3: NA = S1^T @ TH_r (overwrites G), ND = S2^T @ PH_r (overwrites G2)
    gemm_f32_tn32<<<gBack, blk, 0, stream>>>(S1, sS, IC, TH, actS, M2, NA, actS, M2, IC, 1.0f);
    gemm_f32_tn32<<<gBack, blk, 0, stream>>>(S2, sS, IC, PH, actS, M2, ND, actS, M2, IC, 1.0f);

    // Stage 4: output projections + bias + residual; NA/ND reinterpret as [2048,256]
    gemm_f32_nt32<<<gLin, blk, 0, stream>>>(ND, actS, C, Q_w, 0, C, outDet, actS, C, Q_b, detect, actS, C, 1, 1.0f);
    gemm_f32_nt32<<<gLin, blk, 0, stream>>>(NA, actS, C, W_w, 0, C, outAim, actS, C, W_b, aim,    actS, C, 1, 1.0f);
}